// SAGE_76287209112086
// MI455X (gfx1250) — compile-verified
//
#include <hip/hip_runtime.h>
#include <hip/hip_bf16.h>

typedef float v2f __attribute__((ext_vector_type(2)));
typedef float v8f __attribute__((ext_vector_type(8)));

// ---------------------------------------------------------------------------
// Degree / inverse degree
// ---------------------------------------------------------------------------
__global__ void degree_kernel(const int* __restrict__ dst, float* __restrict__ deg, int E) {
    int e = blockIdx.x * blockDim.x + threadIdx.x;
    if (e < E) atomicAdd(&deg[dst[e]], 1.0f);
}

__global__ void deginv_kernel(float* __restrict__ deg, int N) {
    int i = blockIdx.x * blockDim.x + threadIdx.x;
    if (i < N) {
        float d = deg[i];
        deg[i] = (d > 0.0f) ? (1.0f / d) : 0.0f;
    }
}

// ---------------------------------------------------------------------------
// Edge scatter-add: aggr[dst] += feat[src], D=64 features.
// 16 threads per edge, each moves one float4 (coalesced 256B per edge).
// aggr (20.5MB) fits in the 192MB L2 -> atomics resolve on-chip.
// ---------------------------------------------------------------------------
__global__ __launch_bounds__(256) void scatter_add64(
    const float* __restrict__ feat, float* __restrict__ aggr,
    const int* __restrict__ src, const int* __restrict__ dst, int E) {
    int tid = blockIdx.x * blockDim.x + threadIdx.x;
    int e = tid >> 4;
    int c = tid & 15;
    if (e >= E) return;
    int s = src[e];
    int d = dst[e];
    const float4 v = ((const float4*)(feat + (size_t)s * 64))[c];
    float* o = aggr + (size_t)d * 64 + c * 4;
    atomicAdd(o + 0, v.x);
    atomicAdd(o + 1, v.y);
    atomicAdd(o + 2, v.z);
    atomicAdd(o + 3, v.w);
}

// ---------------------------------------------------------------------------
// Fused SAGE linear:  out = act( (aggr*deg_inv) @ Wl^T + bl + h @ Wr^T + br )
// One wave computes a 16-row tile across all output columns with
// V_WMMA_F32_16X16X4_F32 (fp32 precision). The K=128 virtual GEMM
// (A = [aggr*dinv | h], B = [Wl^T ; Wr^T]) is split into two K=64 phases so
// every fragment load is a global_load_b64 at a constant immediate offset
// from a fixed per-lane base (no per-iteration address math / cndmasks).
//
// A layout (32-bit 16x4): lanes 0-15 -> M=lane, K={k0,k0+1};
//                         lanes 16-31 -> M=lane-16, K={k0+2,k0+3}.
// B layout mirrors with N=lane&15. C/D: VGPR r -> row r (+8 for lanes>=16).
// ---------------------------------------------------------------------------
template <int DOUT, bool RELU>
__global__ void __launch_bounds__(256) sage_gemm_kernel(
    const float* __restrict__ aggr, const float* __restrict__ h,
    const float* __restrict__ deg_inv,
    const float* __restrict__ Wl, const float* __restrict__ bl,
    const float* __restrict__ Wr, const float* __restrict__ br,
    float* __restrict__ out, int nTiles) {
    constexpr int NT = (DOUT + 15) / 16;  // column tiles

    const int lane = threadIdx.x & 31;
    const int wave = threadIdx.x >> 5;
    const int rowTile = blockIdx.x * 8 + wave;
    if (rowTile >= nTiles) return;  // wave-uniform: EXEC stays all-ones for WMMA

    const int rowBase = rowTile * 16;
    const int mloc = lane & 15;
    const int khalf = (lane >> 4) * 2;  // lanes 16-31 hold K+2,K+3
    const int m = rowBase + mloc;

    const float dsc = deg_inv[m];
    // Fixed per-lane bases; all 8-byte aligned (khalf even, rows 256B).
    const float* __restrict__ arow = aggr + (size_t)m * 64 + khalf;
    const float* __restrict__ hrow = h + (size_t)m * 64 + khalf;

    const float* wlB[NT];
    const float* wrB[NT];
#pragma unroll
    for (int t = 0; t < NT; ++t) {
        int n = t * 16 + mloc;
        int ncl = (n < DOUT) ? n : (DOUT - 1);  // clamp pad cols (stores masked)
        wlB[t] = Wl + (size_t)ncl * 64 + khalf;
        wrB[t] = Wr + (size_t)ncl * 64 + khalf;
    }

    v8f acc[NT] = {};

    // Phase 1: (aggr * deg_inv) @ Wl^T
#pragma unroll
    for (int k0 = 0; k0 < 64; k0 += 4) {
        v2f a = *(const v2f*)(arow + k0);
        a.x *= dsc;
        a.y *= dsc;
#pragma unroll
        for (int t = 0; t < NT; ++t) {
            v2f b = *(const v2f*)(wlB[t] + k0);
            acc[t] = __builtin_amdgcn_wmma_f32_16x16x4_f32(
                false, a, false, b, (short)0, acc[t], false, false);
        }
    }

    // Phase 2: h @ Wr^T
#pragma unroll
    for (int k0 = 0; k0 < 64; k0 += 4) {
        v2f a = *(const v2f*)(hrow + k0);
#pragma unroll
        for (int t = 0; t < NT; ++t) {
            v2f b = *(const v2f*)(wrB[t] + k0);
            acc[t] = __builtin_amdgcn_wmma_f32_16x16x4_f32(
                false, a, false, b, (short)0, acc[t], false, false);
        }
    }

    // Bias + activation + store.
    const int mrow0 = rowBase + ((lane < 16) ? 0 : 8);
#pragma unroll
    for (int t = 0; t < NT; ++t) {
        int n = t * 16 + mloc;
        if (n < DOUT) {
            float bias = bl[n] + br[n];
#pragma unroll
            for (int r = 0; r < 8; ++r) {
                float v = acc[t][r] + bias;
                if (RELU) v = fmaxf(v, 0.0f);
                out[(size_t)(mrow0 + r) * DOUT + n] = v;
            }
        }
    }
}

// ---------------------------------------------------------------------------
// Per-node masked NLL of log_softmax over 40 classes, block-reduced.
// sums[0] += nll*mask, sums[1] += mask
// ---------------------------------------------------------------------------
__global__ __launch_bounds__(256) void loss_kernel(
    const float* __restrict__ logits, const int* __restrict__ y,
    const unsigned char* __restrict__ mask, float* __restrict__ sums, int N) {
    int i = blockIdx.x * blockDim.x + threadIdx.x;
    float nll = 0.0f, mm = 0.0f;
    if (i < N && mask[i]) {
        const float* r = logits + (size_t)i * 40;
        float mx = r[0];
#pragma unroll
        for (int j = 1; j < 40; ++j) mx = fmaxf(mx, r[j]);
        float s = 0.0f;
#pragma unroll
        for (int j = 0; j < 40; ++j) s += expf(r[j] - mx);
        float lse = mx + logf(s);
        int cls = y[i];
        nll = lse - r[cls];
        mm = 1.0f;
    }
    __shared__ float sn[256];
    __shared__ float sm[256];
    sn[threadIdx.x] = nll;
    sm[threadIdx.x] = mm;
    __syncthreads();
    for (int o = 128; o > 0; o >>= 1) {
        if ((int)threadIdx.x < o) {
            sn[threadIdx.x] += sn[threadIdx.x + o];
            sm[threadIdx.x] += sm[threadIdx.x + o];
        }
        __syncthreads();
    }
    if (threadIdx.x == 0) {
        atomicAdd(&sums[0], sn[0]);
        atomicAdd(&sums[1], sm[0]);
    }
}

__global__ void finalize_kernel(const float* __restrict__ sums, float* __restrict__ out) {
    out[0] = sums[0] / fmaxf(sums[1], 1.0f);
}

// ---------------------------------------------------------------------------
extern "C" void kernel_launch(void* const* d_in, const int* in_sizes, int n_in,
                              void* d_out, int out_size, void* d_ws, size_t ws_size,
                              hipStream_t stream) {
    const float* x   = (const float*)d_in[0];
    const float* Wl0 = (const float*)d_in[1];
    const float* bl0 = (const float*)d_in[2];
    const float* Wr0 = (const float*)d_in[3];
    const float* br0 = (const float*)d_in[4];
    const float* Wl1 = (const float*)d_in[5];
    const float* bl1 = (const float*)d_in[6];
    const float* Wr1 = (const float*)d_in[7];
    const float* br1 = (const float*)d_in[8];
    const float* Wl2 = (const float*)d_in[9];
    const float* bl2 = (const float*)d_in[10];
    const float* Wr2 = (const float*)d_in[11];
    const float* br2 = (const float*)d_in[12];
    const int* ei    = (const int*)d_in[13];
    const int* y     = (const int*)d_in[14];
    const unsigned char* mask = (const unsigned char*)d_in[15];

    const int E = in_sizes[13] / 2;
    const int N = in_sizes[14];
    const int* dst = ei;        // edge_index[0]
    const int* src = ei + E;    // edge_index[1]

    auto align256 = [](size_t v) { return (v + 255) & ~(size_t)255; };
    char* ws = (char*)d_ws;
    size_t off = 0;
    float* deg   = (float*)(ws + off); off += align256((size_t)N * 4);
    float* sums  = (float*)(ws + off); off += 256;
    float* aggr  = (float*)(ws + off); off += align256((size_t)N * 64 * 4);
    float* hbuf1 = (float*)(ws + off); off += align256((size_t)N * 64 * 4);
    float* hbuf2 = (float*)(ws + off); off += align256((size_t)N * 64 * 4);
    // logits (N x 40) reuse hbuf1 after layer 2 no longer needs it
    float* logits = hbuf1;

    const int nTiles = N / 16;                    // N=80000 -> 5000 tiles
    const dim3 blk(256);
    const int gDeg   = (E + 255) / 256;
    const int gN     = (N + 255) / 256;
    const int gScat  = (int)(((size_t)E * 16 + 255) / 256);
    const int gGemm  = (nTiles + 7) / 8;

    // Degree + inverse
    hipMemsetAsync(deg, 0, (size_t)N * 4, stream);
    hipMemsetAsync(sums, 0, 8, stream);
    degree_kernel<<<gDeg, blk, 0, stream>>>(dst, deg, E);
    deginv_kernel<<<gN, blk, 0, stream>>>(deg, N);

    // Layer 0: x -> hbuf1 (relu)
    hipMemsetAsync(aggr, 0, (size_t)N * 64 * 4, stream);
    scatter_add64<<<gScat, blk, 0, stream>>>(x, aggr, src, dst, E);
    sage_gemm_kernel<64, true><<<gGemm, blk, 0, stream>>>(
        aggr, x, deg, Wl0, bl0, Wr0, br0, hbuf1, nTiles);

    // Layer 1: hbuf1 -> hbuf2 (relu)
    hipMemsetAsync(aggr, 0, (size_t)N * 64 * 4, stream);
    scatter_add64<<<gScat, blk, 0, stream>>>(hbuf1, aggr, src, dst, E);
    sage_gemm_kernel<64, true><<<gGemm, blk, 0, stream>>>(
        aggr, hbuf1, deg, Wl1, bl1, Wr1, br1, hbuf2, nTiles);

    // Layer 2: hbuf2 -> logits (no relu, 40 cols)
    hipMemsetAsync(aggr, 0, (size_t)N * 64 * 4, stream);
    scatter_add64<<<gScat, blk, 0, stream>>>(hbuf2, aggr, src, dst, E);
    sage_gemm_kernel<40, false><<<gGemm, blk, 0, stream>>>(
        aggr, hbuf2, deg, Wl2, bl2, Wr2, br2, logits, nTiles);

    // Loss
    loss_kernel<<<gN, blk, 0, stream>>>(logits, y, mask, sums, N);
    finalize_kernel<<<1, 1, 0, stream>>>(sums, (float*)d_out);
}